// Entropy_70600672411748
// MI455X (gfx1250) — compile-verified
//
#include <hip/hip_runtime.h>
#include <cstdint>

#define EPS 1e-6f

namespace {
constexpr int H = 256, W = 256;          // input plane
constexpr int HO = 255, WO = 255;        // output plane (k=2, stride 1)
constexpr int ROWS = 16;                 // output rows per tile
constexpr int THREADS = 256;             // 8 wave32
constexpr int TILES_PER_PLANE = (HO + ROWS - 1) / ROWS;  // 16
}

// LDS (addrspace 3) pointer type: its integer value is the LDS byte offset,
// which is exactly what GLOBAL_LOAD_ASYNC_TO_LDS wants in its VDST VGPR.
typedef __attribute__((address_space(3))) float* lds_fp;

__global__ __launch_bounds__(THREADS) void entropy2x2_kernel(
    const float* __restrict__ x, float* __restrict__ out)
{
    // 17 rows x 256 floats = 17 KB of the 320 KB/WGP LDS.
    __shared__ float smem[(ROWS + 1) * W];

    const int plane = blockIdx.x / TILES_PER_PLANE;
    const int tile  = blockIdx.x % TILES_PER_PLANE;
    const int r0       = tile * ROWS;
    const int rows_out = (HO - r0) < ROWS ? (HO - r0) : ROWS;  // 16 (15 for last tile)
    const int rows_in  = rows_out + 1;                         // input rows needed

    const float* src = x + (size_t)plane * (H * W);

    const int tid = threadIdx.x;

    // ---- Stage input patch (rows_in x 256 fp32) into LDS via async copy ----
    // Each lane moves one 16-byte chunk per iteration; tracked on ASYNCcnt,
    // bypasses VGPRs entirely.
    const unsigned lds_base = (unsigned)(uintptr_t)(lds_fp)&smem[0];
    const int n_chunks = rows_in * (W / 4);   // 64 x B128 chunks per row

    for (int c = tid; c < n_chunks; c += THREADS) {
        unsigned lds_off = lds_base + (unsigned)c * 16u;
        unsigned goff    = (unsigned)(r0 * W) * 4u + (unsigned)c * 16u;
        asm volatile("global_load_async_to_lds_b128 %0, %1, %2"
                     :
                     : "v"(lds_off), "v"(goff), "s"(src)
                     : "memory");
    }
    // Drain this wave's async copies, then sync the workgroup so all waves
    // see the full LDS tile.
    asm volatile("s_wait_asynccnt 0x0" ::: "memory");
    __syncthreads();

    // ---- Compute: one lane per output column, loop over 16 rows ----
    const int j = tid;                 // output column 0..254 (lane 255 idles)
    if (j < WO) {
        float* orow = out + (size_t)plane * (HO * WO) + (size_t)r0 * WO + j;
        #pragma unroll 4
        for (int r = 0; r < rows_out; ++r) {
            // Conflict-free: consecutive lanes read consecutive LDS banks.
            float w0 = smem[r * W + j];
            float w1 = smem[r * W + j + 1];
            float w2 = smem[(r + 1) * W + j];
            float w3 = smem[(r + 1) * W + j + 1];

            float s   = w0 + w1 + w2 + w3 + EPS;
            float inv = __builtin_amdgcn_rcpf(s);      // v_rcp_f32
            float p0 = w0 * inv, p1 = w1 * inv, p2 = w2 * inv, p3 = w3 * inv;
            float e = p0 * __logf(p0 + EPS)            // v_log_f32 path
                    + p1 * __logf(p1 + EPS)
                    + p2 * __logf(p2 + EPS)
                    + p3 * __logf(p3 + EPS);
            // Output is write-once: non-temporal store keeps L2 for the reads.
            __builtin_nontemporal_store(-e, orow + (size_t)r * WO);
        }
    }
}

extern "C" void kernel_launch(void* const* d_in, const int* in_sizes, int n_in,
                              void* d_out, int out_size, void* d_ws, size_t ws_size,
                              hipStream_t stream) {
    (void)n_in; (void)out_size; (void)d_ws; (void)ws_size;
    const float* x = (const float*)d_in[0];
    float* out = (float*)d_out;

    const int planes = in_sizes[0] / (H * W);   // 8*64 = 512
    dim3 grid(planes * TILES_PER_PLANE);        // 8192 blocks
    entropy2x2_kernel<<<grid, dim3(THREADS), 0, stream>>>(x, out);
}